// Eq1to3_75350906241866
// MI455X (gfx1250) — compile-verified
//
#include <hip/hip_runtime.h>

typedef float v2f __attribute__((ext_vector_type(2)));
typedef float v4f __attribute__((ext_vector_type(4)));
typedef float v8f __attribute__((ext_vector_type(8)));

constexpr int kInDim  = 16;
constexpr int kOutDim = 16;
constexpr int kN      = 4;
constexpr int kM      = 96;

// Workspace layout (floats): Y[n][b][s][i] then Sb[n][s] (= S + bias)
constexpr int kYOff = 0;
constexpr int kSOff = kN * 3 * kOutDim * kM;   // 18432 floats

// ---------------------------------------------------------------------------
// Phase 1: per-n tiny GEMMs via V_WMMA_F32_16X16X4_F32.
//   Y[b][s][i] = sum_d coefs[d,s,b] * x[n,d,i]   (b = 0..2)
//   Sb[s]      = sum_d (sum_i x[n,d,i]) * coefs[d,s,3] + bias[s]
// One wave (32 lanes) per block, one block per n. EXEC is all-ones as WMMA
// requires. K=16 is accumulated as four chained K=4 WMMAs.
// ---------------------------------------------------------------------------
__global__ __launch_bounds__(32) void eq1to3_phase1(
    const float* __restrict__ x,       // (4,16,96)
    const float* __restrict__ coefs,   // (16,16,4)
    const float* __restrict__ bias,    // 16
    float* __restrict__ ws)
{
    const int n    = blockIdx.x;
    const int lane = threadIdx.x;      // 0..31
    const int mrow = lane & 15;        // M (or N) index within half-wave
    const int khi  = lane >> 4;        // half-wave selector

    const float* xn = x + n * (kInDim * kM);
    float* Yw = ws + kYOff;
    float* Sw = ws + kSOff;

    for (int bb = 0; bb < 3; ++bb) {
        for (int it = 0; it < kM / 16; ++it) {
            v8f acc = {};
            #pragma unroll
            for (int kk = 0; kk < 4; ++kk) {
                const int k0 = 4 * kk + 2 * khi;
                // A (16x4) = coefs[d, s, bb] with M=s (lane), K=d chunk.
                v2f a;
                a.x = coefs[(k0 + 0) * (kOutDim * 4) + mrow * 4 + bb];
                a.y = coefs[(k0 + 1) * (kOutDim * 4) + mrow * 4 + bb];
                // B (4x16) = x[n, d, it*16 + n'] with K=d chunk, N=lane.
                v2f bmat;
                bmat.x = xn[(k0 + 0) * kM + it * 16 + mrow];
                bmat.y = xn[(k0 + 1) * kM + it * 16 + mrow];
                acc = __builtin_amdgcn_wmma_f32_16x16x4_f32(
                    /*neg_a=*/false, a, /*neg_b=*/false, bmat,
                    /*c_mod=*/(short)0, acc, /*reuse_a=*/false, /*reuse_b=*/false);
            }
            // D layout: vgpr v, lanes 0-15 -> s=v, lanes 16-31 -> s=v+8; N=lane.
            #pragma unroll
            for (int v = 0; v < 8; ++v) {
                const int srow = v + 8 * khi;
                Yw[((n * 3 + bb) * kOutDim + srow) * kM + it * 16 + mrow] = acc[v];
            }
        }
    }

    // S term + bias.
    __shared__ float xsum[kInDim];
    if (lane < kInDim) {
        float s = 0.f;
        const float* row = xn + lane * kM;
        for (int i = 0; i < kM; ++i) s += row[i];
        xsum[lane] = s;
    }
    __syncthreads();
    if (lane < kOutDim) {
        float s = 0.f;
        #pragma unroll
        for (int d = 0; d < kInDim; ++d)
            s += xsum[d] * coefs[d * (kOutDim * 4) + lane * 4 + 3];
        Sw[n * kOutDim + lane] = s + bias[lane];
    }
}

// ---------------------------------------------------------------------------
// Phase 2: stream 226.5 MB of output. One block per (n,s,i) slab of 96x96
// floats. Contiguous float4 non-temporal stores (output >> 192 MB L2 and is
// write-once, so bypass temporal caching).
// ---------------------------------------------------------------------------
__global__ __launch_bounds__(256) void eq1to3_phase2(
    const float* __restrict__ ws, float* __restrict__ out)
{
    const int bid = blockIdx.x;            // = (n*16 + s)*96 + i
    const int i  = bid % kM;
    const int ns = bid / kM;
    const int s  = ns % kOutDim;
    const int n  = ns / kOutDim;

    const float* Yw = ws + kYOff;
    const float* Sw = ws + kSOff;

    __shared__ float y1[kM];
    __shared__ float y2[kM];
    const int tid = threadIdx.x;
    const float* Y1g = Yw + ((n * 3 + 1) * kOutDim + s) * kM;
    const float* Y2g = Yw + ((n * 3 + 2) * kOutDim + s) * kM;
    if (tid < kM)            y1[tid]       = Y1g[tid];
    else if (tid < 2 * kM)   y2[tid - kM]  = Y2g[tid - kM];
    __syncthreads();

    const float base =
        Yw[((n * 3 + 0) * kOutDim + s) * kM + i] + Sw[n * kOutDim + s];

    float* orow = out + (size_t)bid * (kM * kM);
    constexpr int kQuads = (kM * kM) / 4;   // 2304 float4s per slab
    for (int idx = tid; idx < kQuads; idx += 256) {
        const int j  = idx / (kM / 4);
        const int k4 = idx % (kM / 4);
        const float bj = base + y1[j];
        v4f v;
        v.x = bj + y2[4 * k4 + 0];
        v.y = bj + y2[4 * k4 + 1];
        v.z = bj + y2[4 * k4 + 2];
        v.w = bj + y2[4 * k4 + 3];
        __builtin_nontemporal_store(v, (v4f*)(orow + (size_t)idx * 4));
    }
}

extern "C" void kernel_launch(void* const* d_in, const int* in_sizes, int n_in,
                              void* d_out, int out_size, void* d_ws, size_t ws_size,
                              hipStream_t stream) {
    const float* x     = (const float*)d_in[0];
    const float* coefs = (const float*)d_in[1];
    const float* bias  = (const float*)d_in[2];
    float* out = (float*)d_out;
    float* ws  = (float*)d_ws;

    eq1to3_phase1<<<kN, 32, 0, stream>>>(x, coefs, bias, ws);
    eq1to3_phase2<<<kN * kOutDim * kM, 256, 0, stream>>>(ws, out);
}